// DynamicMemory_39075612459450
// MI455X (gfx1250) — compile-verified
//
#include <hip/hip_runtime.h>
#include <math.h>

typedef __attribute__((ext_vector_type(2))) float v2f;
typedef __attribute__((ext_vector_type(8))) float v8f;

#define HDIM 256
#define NSLOT 64
#define BATCH 4096

// ---------------------------------------------------------------------------
// K1: Acand[n,k] = sum_h h[n,h]*U[k,h] + w[n,h]*V[k,h]   (64x256 output)
// One wave (32 lanes) per 16x16 tile, K=256 in steps of 4, f32 WMMA.
// ---------------------------------------------------------------------------
__global__ __launch_bounds__(32) void cand_gemm_wmma(
    const float* __restrict__ h, const float* __restrict__ w,
    const float* __restrict__ U, const float* __restrict__ V,
    float* __restrict__ Acand) {
  const int lane = threadIdx.x;
  const int ll = lane & 15;        // M / N within tile
  const int lh = lane >> 4;        // selects K pair {0,1} vs {2,3}
  const int mbase = blockIdx.x * 16;   // over N=64 rows of h
  const int cbase = blockIdx.y * 16;   // over H=256 output cols
  v8f c = {};
#pragma unroll 4
  for (int kk = 0; kk < HDIM / 4; ++kk) {
    const int col = kk * 4 + lh * 2;
    v2f ah = *(const v2f*)(h + (mbase + ll) * HDIM + col);
    v2f bu = *(const v2f*)(U + (cbase + ll) * HDIM + col);
    c = __builtin_amdgcn_wmma_f32_16x16x4_f32(false, ah, false, bu,
                                              (short)0, c, false, false);
    v2f aw = *(const v2f*)(w + (mbase + ll) * HDIM + col);
    v2f bv = *(const v2f*)(V + (cbase + ll) * HDIM + col);
    c = __builtin_amdgcn_wmma_f32_16x16x4_f32(false, aw, false, bv,
                                              (short)0, c, false, false);
  }
#pragma unroll
  for (int r = 0; r < 8; ++r)
    Acand[(mbase + r + 8 * lh) * HDIM + cbase + ll] = c[r];
}

// ---------------------------------------------------------------------------
// K2: SW[b,k] = s_t[b,:] . W[k,:]        (grid.y tiles 0..15)
//     G [b,n] = s_t[b,:] . (h+w)[n,:]    (grid.y tiles 16..19)
// One wave per 16x16 output tile, K=256.
// ---------------------------------------------------------------------------
__global__ __launch_bounds__(32) void st_gemm_wmma(
    const float* __restrict__ s_t, const float* __restrict__ Wm,
    const float* __restrict__ h, const float* __restrict__ w,
    float* __restrict__ SW, float* __restrict__ G) {
  const int lane = threadIdx.x;
  const int ll = lane & 15;
  const int lh = lane >> 4;
  const int mbase = blockIdx.x * 16;   // over B=4096 rows
  const int y = blockIdx.y;
  v8f c = {};
  if (y < 16) {
    const int cbase = y * 16;
#pragma unroll 4
    for (int kk = 0; kk < HDIM / 4; ++kk) {
      const int col = kk * 4 + lh * 2;
      v2f a = *(const v2f*)(s_t + (mbase + ll) * HDIM + col);
      v2f b = *(const v2f*)(Wm + (cbase + ll) * HDIM + col);
      c = __builtin_amdgcn_wmma_f32_16x16x4_f32(false, a, false, b,
                                                (short)0, c, false, false);
    }
#pragma unroll
    for (int r = 0; r < 8; ++r)
      SW[(mbase + r + 8 * lh) * HDIM + cbase + ll] = c[r];
  } else {
    const int cbase = (y - 16) * 16;   // over N=64 gate cols
#pragma unroll 4
    for (int kk = 0; kk < HDIM / 4; ++kk) {
      const int col = kk * 4 + lh * 2;
      v2f a = *(const v2f*)(s_t + (mbase + ll) * HDIM + col);
      v2f bh = *(const v2f*)(h + (cbase + ll) * HDIM + col);
      v2f bw = *(const v2f*)(w + (cbase + ll) * HDIM + col);
      v2f b = bh + bw;                 // gate uses (h + w)
      c = __builtin_amdgcn_wmma_f32_16x16x4_f32(false, a, false, b,
                                                (short)0, c, false, false);
    }
#pragma unroll
    for (int r = 0; r < 8; ++r)
      G[(mbase + r + 8 * lh) * NSLOT + cbase + ll] = c[r];
  }
}

// ---------------------------------------------------------------------------
// K3: out[b,n,:] = normalize( h[n,:] + sigmoid(G[b,n]) *
//                             prelu(Acand[n,:] + SW[b,:]) )
// One block per batch row b; SW row + gates cached in LDS; each wave32 owns
// whole (b,n) rows -> sum-of-squares reduction is pure in-wave shuffles.
// Output stores are non-temporal: 256 MB one-shot stream > 192 MB L2, so
// keeping it out of cache preserves the reused h/Acand tables in L2.
// ---------------------------------------------------------------------------
__global__ __launch_bounds__(256) void fuse_norm(
    const float* __restrict__ h, const float* __restrict__ Acand,
    const float* __restrict__ SW, const float* __restrict__ G,
    const float* __restrict__ prelu_a, float* __restrict__ out) {
  __shared__ float sw_s[HDIM];
  __shared__ float g_s[NSLOT];
  const int b = blockIdx.x;
  const int t = threadIdx.x;
  sw_s[t] = SW[(size_t)b * HDIM + t];
  if (t < NSLOT) {
    const float x = G[(size_t)b * NSLOT + t];
    g_s[t] = 1.0f / (1.0f + expf(-x));
  }
  __syncthreads();
  const float a = prelu_a[0];
  const int wave = t >> 5;
  const int lane = t & 31;
#pragma unroll
  for (int nn = 0; nn < NSLOT / 8; ++nn) {
    const int n = wave + nn * 8;
    const float g = g_s[n];
    float vals[8];
    float ss = 0.0f;
#pragma unroll
    for (int i = 0; i < 8; ++i) {
      const int k = i * 32 + lane;              // coalesced 128B per wave
      float x = Acand[n * HDIM + k] + sw_s[k];
      x = (x >= 0.0f) ? x : a * x;              // PReLU
      const float v = h[n * HDIM + k] + g * x;
      vals[i] = v;
      ss += v * v;
    }
#pragma unroll
    for (int off = 16; off > 0; off >>= 1)      // wave32 reduction
      ss += __shfl_xor(ss, off, 32);
    const float inv = rsqrtf(ss);
    float* op = out + ((size_t)b * NSLOT + n) * HDIM;
#pragma unroll
    for (int i = 0; i < 8; ++i)
      __builtin_nontemporal_store(vals[i] * inv, op + i * 32 + lane);
  }
}

// ---------------------------------------------------------------------------
extern "C" void kernel_launch(void* const* d_in, const int* in_sizes, int n_in,
                              void* d_out, int out_size, void* d_ws,
                              size_t ws_size, hipStream_t stream) {
  const float* s_t = (const float*)d_in[0];   // [B,H]
  const float* h   = (const float*)d_in[1];   // [1,N,H]
  const float* w   = (const float*)d_in[2];   // [1,N,H]
  const float* U   = (const float*)d_in[3];   // [H,H]
  const float* V   = (const float*)d_in[4];   // [H,H]
  const float* Wm  = (const float*)d_in[5];   // [H,H]
  const float* pa  = (const float*)d_in[6];   // [1]
  float* out = (float*)d_out;

  char* ws = (char*)d_ws;
  float* SW = (float*)ws;                                          // 4 MB
  float* G  = (float*)(ws + (size_t)BATCH * HDIM * sizeof(float)); // 1 MB
  float* Ac = (float*)(ws + (size_t)BATCH * HDIM * sizeof(float)
                          + (size_t)BATCH * NSLOT * sizeof(float)); // 64 KB

  // Batch-invariant candidate table (tiny, WMMA).
  cand_gemm_wmma<<<dim3(NSLOT / 16, HDIM / 16), 32, 0, stream>>>(h, w, U, V, Ac);
  // Batch GEMMs: SW = s_t @ W^T and gate logits G = s_t @ (h+w)^T (WMMA).
  st_gemm_wmma<<<dim3(BATCH / 16, 20), 32, 0, stream>>>(s_t, Wm, h, w, SW, G);
  // Bandwidth-bound fused epilogue with NT output stream.
  fuse_norm<<<BATCH, 256, 0, stream>>>(h, Ac, SW, G, pa, out);
}